// Bloom_16226386444301
// MI455X (gfx1250) — compile-verified
//
#include <hip/hip_runtime.h>
#include <hip/hip_bf16.h>

typedef float v2f __attribute__((ext_vector_type(2)));
typedef float v8f __attribute__((ext_vector_type(8)));

#define NSIG   2048
#define NC     36            // (LMAX+1)^2
#define RES    200
#define TOPK   16
#define TILE_N 16            // signals per block (WMMA M)
#define NW     8             // wave32s per block
#define NTHR   (NW * 32)
#define AW     208           // padded alpha width (13 * 16)
#define NTILE  13            // alpha tiles of 16 covering 200
#define ROWS_PER_CHUNK 25
#define NCHUNK (RES / ROWS_PER_CHUNK)   // 8 beta chunks -> 128*8 = 1024 blocks
#define CAP    512           // max candidates kept per signal
#define NEGF   (-3.0e38f)

__global__ void Bloom_zero_counts(int* gcnt) {
    int i = blockIdx.x * blockDim.x + threadIdx.x;
    if (i < NSIG) gcnt[i] = 0;
}

// Fused: f32 WMMA GEMM (16 signals x 200x200 grid) -> LDS ring -> peak detect -> candidate emit.
__global__ __launch_bounds__(NTHR)
void Bloom_sh_peak_kernel(const float* __restrict__ signal,
                          const float* __restrict__ Y,
                          const float* __restrict__ minrp,
                          int*   __restrict__ gcnt,
                          float* __restrict__ gval,
                          int*   __restrict__ gidx) {
    __shared__ float sA[TILE_N][NC + 4];      // signal tile, padded
    __shared__ float ring[3][TILE_N][AW];     // 3 beta rows of f

    const int tid  = threadIdx.x;
    const int lane = tid & 31;
    const int wid  = tid >> 5;
    const int n0   = blockIdx.x * TILE_N;
    const int R0   = blockIdx.y * ROWS_PER_CHUNK;
    const int R1   = R0 + ROWS_PER_CHUNK - 1;
    const float minr = minrp[0];

    // Load 16x36 signal tile into LDS.
    for (int i = tid; i < TILE_N * NC; i += NTHR)
        sA[i / NC][i % NC] = signal[(size_t)(n0 + i / NC) * NC + (i % NC)];
    __syncthreads();

    // Per-lane A fragments for V_WMMA_F32_16X16X4_F32.
    // ISA layout (16x4 f32 A): lanes 0-15 -> M=lane, VGPR0/1 = K0/K1;
    // lanes 16-31 -> M=lane-16, VGPR0/1 = K2/K3.
    const int am_   = lane & 15;
    const int khalf = (lane >> 4) * 2;
    v2f aF[9];
#pragma unroll
    for (int kk = 0; kk < 9; ++kk) {
        const int kb = kk * 4 + khalf;
        aF[kk].x = sA[am_][kb];
        aF[kk].y = sA[am_][kb + 1];
    }
    const int mbase = (lane >> 4) * 8;   // C/D layout: VGPR r -> M = r + mbase

    const int b_start = R0 - 1;
    const int b_end   = R1 + 1;
    for (int b = b_start; b <= b_end; ++b) {
        const int s = (b - b_start) % 3;
        if (b >= 0 && b < RES) {
            // Compute f row b: 13 alpha tiles of 16, one 16x16 tile per wave-iteration.
            for (int t = wid; t < NTILE; t += NW) {
                const int a0 = t * 16;
                const int ac = min(a0 + am_, RES - 1);       // clamp pad lanes
                const float* yrow = Y + (size_t)(b * RES + ac) * NC;
                v8f c = {0.f, 0.f, 0.f, 0.f, 0.f, 0.f, 0.f, 0.f};
#pragma unroll
                for (int kk = 0; kk < 9; ++kk) {
                    const int kb = kk * 4 + khalf;           // B mirrors A's K half-split
                    v2f bv;
                    bv.x = yrow[kb];
                    bv.y = yrow[kb + 1];
                    c = __builtin_amdgcn_wmma_f32_16x16x4_f32(
                            false, aF[kk], false, bv, (short)0, c, false, false);
                }
#pragma unroll
                for (int r = 0; r < 8; ++r)
                    ring[s][mbase + r][a0 + am_] = c[r];
            }
        } else {
            // Virtual -inf row at the beta poles.
            for (int i = tid; i < TILE_N * AW; i += NTHR)
                ring[s][i / AW][i % AW] = NEGF;
        }
        __syncthreads();

        // Detect peaks on row r = b-1 (needs rows r-1, r, r+1 in the ring).
        const int r = b - 1;
        if (r >= R0) {
            const int sp = (b - 2 - b_start) % 3;
            const int sc = (b - 1 - b_start) % 3;
            const int sn = s;
            for (int i = tid; i < TILE_N * RES; i += NTHR) {
                const int m  = i / RES;
                const int a  = i % RES;
                const int al = (a + RES - 1) % RES;          // alpha wraps
                const int ar = (a + 1) % RES;
                const float f = ring[sc][m][a];
                float nmax = fmaxf(ring[sc][m][al], ring[sc][m][ar]);
                nmax = fmaxf(nmax, fmaxf(ring[sp][m][al],
                             fmaxf(ring[sp][m][a], ring[sp][m][ar])));
                nmax = fmaxf(nmax, fmaxf(ring[sn][m][al],
                             fmaxf(ring[sn][m][a], ring[sn][m][ar])));
                if (f > nmax && f > minr) {
                    const int n = n0 + m;
                    const int p = atomicAdd(&gcnt[n], 1);
                    if (p < CAP) {
                        gval[(size_t)n * CAP + p] = f;
                        gidx[(size_t)n * CAP + p] = r * RES + a;
                    }
                }
            }
        }
        __syncthreads();
    }
}

// One wave32 per signal: register-resident top-16 via shfl-xor arg-max.
__global__ __launch_bounds__(256)
void Bloom_select_kernel(const float* __restrict__ signal,
                         const float* __restrict__ pos,
                         const float* __restrict__ dirs,
                         const int*   __restrict__ gcnt,
                         const float* __restrict__ gval,
                         const int*   __restrict__ gidx,
                         float* __restrict__ outPos,
                         int*   __restrict__ outIdx,
                         int*   __restrict__ outMask) {
    const int lane = threadIdx.x & 31;
    const int wid  = threadIdx.x >> 5;
    const int n    = blockIdx.x * 8 + wid;
    if (n >= NSIG) return;

    const int cnt = min(gcnt[n], CAP);
    float v[CAP / 32];
    int   gi[CAP / 32];
#pragma unroll
    for (int t = 0; t < CAP / 32; ++t) {
        const int j = lane + t * 32;
        const bool ok = j < cnt;
        v[t]  = ok ? gval[(size_t)n * CAP + j] : NEGF;
        gi[t] = ok ? gidx[(size_t)n * CAP + j] : 0x7fffffff;
    }
    const float px = pos[(size_t)n * 3 + 0];
    const float py = pos[(size_t)n * 3 + 1];
    const float pz = pos[(size_t)n * 3 + 2];

    for (int k = 0; k < TOPK; ++k) {
        float bv = NEGF;
        int   bg = 0x7fffffff;
#pragma unroll
        for (int t = 0; t < CAP / 32; ++t)
            if (v[t] > bv || (v[t] == bv && gi[t] < bg)) { bv = v[t]; bg = gi[t]; }
        for (int off = 16; off > 0; off >>= 1) {
            const float ov = __shfl_xor(bv, off, 32);
            const int   og = __shfl_xor(bg, off, 32);
            if (ov > bv || (ov == bv && og < bg)) { bv = ov; bg = og; }
        }
        const bool valid = bv > (0.5f * NEGF);
        if (valid) {
#pragma unroll
            for (int t = 0; t < CAP / 32; ++t)
                if (gi[t] == bg) { v[t] = NEGF; gi[t] = 0x7fffffff; }
        }
        if (lane == 0) {
            const size_t o = (size_t)n * TOPK + k;
            if (valid) {
                outPos[o * 3 + 0] = dirs[(size_t)bg * 3 + 0] * bv + px;
                outPos[o * 3 + 1] = dirs[(size_t)bg * 3 + 1] * bv + py;
                outPos[o * 3 + 2] = dirs[(size_t)bg * 3 + 2] * bv + pz;
                outMask[o] = 1;
            } else if (k == 0) {                 // no peaks -> l=1 fallback in slot 0
                outPos[o * 3 + 0] = signal[(size_t)n * NC + 1] + px;
                outPos[o * 3 + 1] = signal[(size_t)n * NC + 2] + py;
                outPos[o * 3 + 2] = signal[(size_t)n * NC + 3] + pz;
                outMask[o] = 1;
            } else {                             // radii=0 -> new_pos = pos, mask 0
                outPos[o * 3 + 0] = px;
                outPos[o * 3 + 1] = py;
                outPos[o * 3 + 2] = pz;
                outMask[o] = 0;
            }
            outIdx[o] = n;                       // broadcast arange(N)
        }
    }
}

extern "C" void kernel_launch(void* const* d_in, const int* in_sizes, int n_in,
                              void* d_out, int out_size, void* d_ws, size_t ws_size,
                              hipStream_t stream) {
    (void)in_sizes; (void)n_in; (void)out_size; (void)ws_size;
    const float* signal = (const float*)d_in[0];
    const float* pos    = (const float*)d_in[1];
    const float* minr   = (const float*)d_in[2];
    const float* Y      = (const float*)d_in[3];
    const float* dirs   = (const float*)d_in[4];

    int*   gcnt = (int*)d_ws;
    float* gval = (float*)((char*)d_ws + NSIG * sizeof(int));
    int*   gidx = (int*)((char*)d_ws + NSIG * sizeof(int)
                         + (size_t)NSIG * CAP * sizeof(float));

    float* outPos  = (float*)d_out;
    int*   outIdx  = (int*)d_out + (size_t)NSIG * TOPK * 3;
    int*   outMask = outIdx + (size_t)NSIG * TOPK;

    Bloom_zero_counts<<<(NSIG + 255) / 256, 256, 0, stream>>>(gcnt);

    dim3 grid(NSIG / TILE_N, NCHUNK);
    Bloom_sh_peak_kernel<<<grid, NTHR, 0, stream>>>(signal, Y, minr, gcnt, gval, gidx);

    Bloom_select_kernel<<<NSIG / 8, 256, 0, stream>>>(signal, pos, dirs,
                                                      gcnt, gval, gidx,
                                                      outPos, outIdx, outMask);
}